// DetectionLoss_12257836663248
// MI455X (gfx1250) — compile-verified
//
#include <hip/hip_runtime.h>
#include <math.h>

// ---------------------------------------------------------------------------
// DetectionLoss for MI455X (gfx1250, wave32).
// Stage 1: fused IoU/match + per-anchor losses, deterministic block partials.
// Stage 2: exact top-k negative-sum via bitwise binary search (replaces sort).
// Stage 3: single-wave finalize; batch reduction done with
//          V_WMMA_F32_16X16X4_F32 (D = A x ones + C row-sum trick).
// ---------------------------------------------------------------------------

typedef __attribute__((ext_vector_type(2))) float v2f;
typedef __attribute__((ext_vector_type(8))) float v8f;

#define NBATCH 64
#define NT     50    // GT boxes per image
#define NCELL  192   // 3 scales * 64 batches
#define TPB    256

__global__ void zero_i32_kernel(int* __restrict__ p, int n) {
  int i = blockIdx.x * blockDim.x + threadIdx.x;
  if (i < n) p[i] = 0;
}

__device__ __forceinline__ float softplusf(float x) {
  // logaddexp(0, x), numerically stable
  return fmaxf(x, 0.0f) + log1pf(expf(-fabsf(x)));
}

// One thread per anchor. Block handles a 256-anchor chunk of one (batch, scale).
__global__ __launch_bounds__(TPB) void stage1_kernel(
    const float* __restrict__ p, const float* __restrict__ anchors,
    const float* __restrict__ boxes, const int* __restrict__ labels,
    float* __restrict__ scores,
    float* __restrict__ pObj, float* __restrict__ pCls, float* __restrict__ pLoc,
    int* __restrict__ accI,
    int H, int W, int N, int nChunks, int cellBase)
{
  __shared__ float sb[NT * 4];
  __shared__ int   slab[NT];
  __shared__ float red[TPB];
  __shared__ int   sPos, sNeg;

  const int tid = threadIdx.x;
  const int b   = blockIdx.x / nChunks;
  const int chk = blockIdx.x - b * nChunks;
  const int n   = chk * TPB + tid;

  if (tid < NT * 4) sb[tid]   = boxes[b * NT * 4 + tid];
  if (tid < NT)     slab[tid] = labels[b * NT + tid];
  if (tid == 0) { sPos = 0; sNeg = 0; }
  __syncthreads();

  float myObj = 0.0f, myCls = 0.0f, myLoc = 0.0f;

  if (n < N) {
    const int HW = H * W;
    const int a  = n / HW;
    const int hw = n - a * HW;
    // p layout: (B, 24, H, W) viewed as (B, A=3, 5+C=8, H, W)
    const size_t base = ((size_t)b * 24 + (size_t)a * 8) * (size_t)HW + (size_t)hw;
    const float pb0 = p[base + (size_t)0 * HW];
    const float pb1 = p[base + (size_t)1 * HW];
    const float pb2 = p[base + (size_t)2 * HW];
    const float pb3 = p[base + (size_t)3 * HW];
    const float pob = p[base + (size_t)4 * HW];
    const float pc0 = p[base + (size_t)5 * HW];
    const float pc1 = p[base + (size_t)6 * HW];
    const float pc2 = p[base + (size_t)7 * HW];

    const float ax1 = anchors[n * 4 + 0];
    const float ay1 = anchors[n * 4 + 1];
    const float ax2 = anchors[n * 4 + 2];
    const float ay2 = anchors[n * 4 + 3];
    const float areaA = (ax2 - ax1) * (ay2 - ay1);

    float best = -1.0e30f;
    int   bidx = 0;
#pragma unroll 5
    for (int t = 0; t < NT; ++t) {
      const float gx1 = sb[t * 4 + 0];
      const float gy1 = sb[t * 4 + 1];
      const float gx2 = sb[t * 4 + 2];
      const float gy2 = sb[t * 4 + 3];
      const float iw = fminf(ax2, gx2) - fmaxf(ax1, gx1);
      const float ih = fminf(ay2, gy2) - fmaxf(ay1, gy1);
      const float inter = fmaxf(iw, 0.0f) * fmaxf(ih, 0.0f);
      const float areaG = (gx2 - gx1) * (gy2 - gy1);
      float iou = inter / (areaA + areaG - inter + 1e-9f);
      if (slab[t] <= 0) iou = -1.0f;          // invalid GT -> masked
      if (iou > best) { best = iou; bidx = t; } // first-occurrence argmax
    }

    const bool pos  = best >= 0.5f;
    const bool neg  = best <  0.4f;
    const float objl = softplusf(pob);          // logaddexp(0, x)

    // hard-negative-mining score: where(neg, obj_loss, -inf)
    scores[(size_t)b * N + n] = neg ? objl : -__builtin_inff();

    if (pos) {
      myObj = objl - pob;                       // logaddexp(0,x) - x
      const int tgt = max(slab[bidx] - 1, 0);
      const float m3  = fmaxf(fmaxf(pc0, pc1), pc2);
      const float lse = m3 + logf(expf(pc0 - m3) + expf(pc1 - m3) + expf(pc2 - m3));
      const float tl  = (tgt == 0) ? pc0 : ((tgt == 1) ? pc1 : pc2);
      myCls = lse - tl;                         // cross-entropy
      const float pb[4] = {pb0, pb1, pb2, pb3};
      float locs = 0.0f;
#pragma unroll
      for (int cdx = 0; cdx < 4; ++cdx) {       // smooth-L1
        const float d  = pb[cdx] - sb[bidx * 4 + cdx];
        const float ad = fabsf(d);
        locs += (ad < 1.0f) ? 0.5f * d * d : ad - 0.5f;
      }
      myLoc = locs;
      atomicAdd(&sPos, 1);                      // int: order-independent
    } else if (neg) {
      atomicAdd(&sNeg, 1);
    }
  }

  // deterministic tree reductions (bit-identical across graph replays)
  float blkObj, blkCls, blkLoc;
  red[tid] = myObj; __syncthreads();
  for (int off = TPB / 2; off > 0; off >>= 1) { if (tid < off) red[tid] += red[tid + off]; __syncthreads(); }
  blkObj = red[0]; __syncthreads();
  red[tid] = myCls; __syncthreads();
  for (int off = TPB / 2; off > 0; off >>= 1) { if (tid < off) red[tid] += red[tid + off]; __syncthreads(); }
  blkCls = red[0]; __syncthreads();
  red[tid] = myLoc; __syncthreads();
  for (int off = TPB / 2; off > 0; off >>= 1) { if (tid < off) red[tid] += red[tid + off]; __syncthreads(); }
  blkLoc = red[0];

  if (tid == 0) {
    const int pi = b * nChunks + chk;
    pObj[pi] = blkObj; pCls[pi] = blkCls; pLoc[pi] = blkLoc;
    const int cell = cellBase + b;
    atomicAdd(&accI[cell], sPos);
    atomicAdd(&accI[NCELL + cell], sNeg);
  }
}

// One block per (batch, scale): reduce stage-1 partials, then exact top-k
// negative sum via binary search on IEEE bit patterns (all scores > 0).
__global__ __launch_bounds__(TPB) void stage2_kernel(
    const float* __restrict__ scores,
    const float* __restrict__ pObj, const float* __restrict__ pCls, const float* __restrict__ pLoc,
    const int* __restrict__ accI,
    float* __restrict__ accF, float* __restrict__ topk,
    int N, int nChunks, int cellBase)
{
  __shared__ float rf[TPB];
  __shared__ int   ri[TPB];

  const int tid  = threadIdx.x;
  const int b    = blockIdx.x;
  const int cell = cellBase + b;
  const float* sc = scores + (size_t)b * N;

  float o = 0.0f, c = 0.0f, l = 0.0f;
  for (int i = tid; i < nChunks; i += TPB) {
    o += pObj[b * nChunks + i];
    c += pCls[b * nChunks + i];
    l += pLoc[b * nChunks + i];
  }
  rf[tid] = o; __syncthreads();
  for (int off = TPB / 2; off > 0; off >>= 1) { if (tid < off) rf[tid] += rf[tid + off]; __syncthreads(); }
  if (tid == 0) accF[cell] = rf[0];
  __syncthreads();
  rf[tid] = c; __syncthreads();
  for (int off = TPB / 2; off > 0; off >>= 1) { if (tid < off) rf[tid] += rf[tid + off]; __syncthreads(); }
  if (tid == 0) accF[NCELL + cell] = rf[0];
  __syncthreads();
  rf[tid] = l; __syncthreads();
  for (int off = TPB / 2; off > 0; off >>= 1) { if (tid < off) rf[tid] += rf[tid + off]; __syncthreads(); }
  if (tid == 0) accF[2 * NCELL + cell] = rf[0];
  __syncthreads();

  const int np = accI[cell];
  const int nn = accI[NCELL + cell];
  const int k  = min(nn, 3 * max(np, 1));
  if (k == 0) { if (tid == 0) topk[cell] = 0.0f; return; }

  // smallest u with count(score > u) < k  ==>  u == k-th largest score
  unsigned lo = 0u, hi = 0x7F800000u;  // [0, +inf)
  while (lo < hi) {
    const unsigned mid = lo + ((hi - lo) >> 1);
    const float t = __uint_as_float(mid);
    int cnt = 0;
    for (int i = tid; i < N; i += TPB) cnt += (sc[i] > t) ? 1 : 0;
    ri[tid] = cnt; __syncthreads();
    for (int off = TPB / 2; off > 0; off >>= 1) { if (tid < off) ri[tid] += ri[tid + off]; __syncthreads(); }
    const int total = ri[0]; __syncthreads();
    if (total < k) hi = mid; else lo = mid + 1u;
  }
  const float v = __uint_as_float(lo);

  int cgt = 0; float sgt = 0.0f;
  for (int i = tid; i < N; i += TPB) {
    const float s = sc[i];
    if (s > v) { ++cgt; sgt += s; }
  }
  ri[tid] = cgt; rf[tid] = sgt; __syncthreads();
  for (int off = TPB / 2; off > 0; off >>= 1) {
    if (tid < off) { ri[tid] += ri[tid + off]; rf[tid] += rf[tid + off]; }
    __syncthreads();
  }
  if (tid == 0) topk[cell] = rf[0] + (float)(k - ri[0]) * v;  // ties at v
}

// Single wave. Per-cell losses -> LDS -> WMMA row-sum reduction (B = ones).
__global__ __launch_bounds__(32) void finalize_kernel(
    const float* __restrict__ accF, const int* __restrict__ accI,
    const float* __restrict__ topk, float* __restrict__ out)
{
  __shared__ float sv[3][NCELL];
  const int lane = threadIdx.x;

  for (int i = lane; i < NCELL; i += 32) {   // 192/32 = 6, no divergence
    const int np = accI[i];
    const int nn = accI[NCELL + i];
    const int k  = min(nn, 3 * max(np, 1));
    sv[0][i] = (accF[i] + topk[i]) / (float)max(np + k, 1);
    sv[1][i] = accF[NCELL + i]     / (float)max(np, 1);
    sv[2][i] = accF[2 * NCELL + i] / (float)max(4 * np, 1);
  }
  __syncthreads();

  // A 16x4 f32 layout: lanes 0-15 hold {K0,K1}, lanes 16-31 hold {K2,K3}.
  const int m  = lane & 15;
  const int kh = (lane >> 4) << 1;           // 0 or 2
  v2f ones; ones.x = 1.0f; ones.y = 1.0f;    // B layout irrelevant: all ones

  float tot[3];
#pragma unroll
  for (int q = 0; q < 3; ++q) {
    v8f acc = {};
#pragma unroll
    for (int ch = 0; ch < 3; ++ch) {         // 3 chunks of 64 values
      v2f a;
      a.x = sv[q][ch * 64 + m * 4 + kh + 0];
      a.y = sv[q][ch * 64 + m * 4 + kh + 1];
      // D[m][n] = sum_k A[m][k] + C[m][n]  (exact fp32 row sums)
      acc = __builtin_amdgcn_wmma_f32_16x16x4_f32(
          false, a, false, ones, (short)0, acc, false, false);
    }
    float s = acc[0] + acc[1] + acc[2] + acc[3] + acc[4] + acc[5] + acc[6] + acc[7];
    s += __shfl_xor(s, 16, 32);              // rows 0-7 (+) rows 8-15
    tot[q] = s;
  }

  if (lane == 0) {
    const float lo = tot[0] / 3.0f;
    const float lc = tot[1] / 3.0f;
    const float ll = tot[2] / 3.0f;
    out[0] = lo + lc + 2.0f * ll;  // OBJ_W*lo + CLS_W*lc + LOC_W*ll
    out[1] = lo;
    out[2] = lc;
    out[3] = ll;
  }
}

extern "C" void kernel_launch(void* const* d_in, const int* in_sizes, int n_in,
                              void* d_out, int out_size, void* d_ws, size_t ws_size,
                              hipStream_t stream) {
  (void)in_sizes; (void)n_in; (void)out_size; (void)ws_size;

  const float* p0 = (const float*)d_in[0];
  const float* p1 = (const float*)d_in[1];
  const float* p2 = (const float*)d_in[2];
  const float* a0 = (const float*)d_in[3];
  const float* a1 = (const float*)d_in[4];
  const float* a2 = (const float*)d_in[5];
  const float* bx = (const float*)d_in[6];
  const int*   lb = (const int*)d_in[7];
  float* out = (float*)d_out;

  const int N0 = 19200, N1 = 4800, N2 = 1200;
  const int C0 = (N0 + TPB - 1) / TPB;  // 75
  const int C1 = (N1 + TPB - 1) / TPB;  // 19
  const int C2 = (N2 + TPB - 1) / TPB;  // 5
  const int NBLK = NBATCH * (C0 + C1 + C2);  // 6336

  // Workspace layout (fp32 words): scores | partials(3x) | accF | accI | topk
  float* wsf = (float*)d_ws;
  float* sc0 = wsf;
  float* sc1 = sc0 + (size_t)NBATCH * N0;
  float* sc2 = sc1 + (size_t)NBATCH * N1;
  float* pObj = sc2 + (size_t)NBATCH * N2;
  float* pCls = pObj + NBLK;
  float* pLoc = pCls + NBLK;
  float* accF = pLoc + NBLK;                 // 3*NCELL floats
  int*   accI = (int*)(accF + 3 * NCELL);    // 2*NCELL ints
  float* topk = (float*)(accI + 2 * NCELL);  // NCELL floats

  const int pB1 = NBATCH * C0;               // partial base, scale 1
  const int pB2 = pB1 + NBATCH * C1;         // partial base, scale 2

  zero_i32_kernel<<<(2 * NCELL + TPB - 1) / TPB, TPB, 0, stream>>>(accI, 2 * NCELL);

  stage1_kernel<<<NBATCH * C0, TPB, 0, stream>>>(p0, a0, bx, lb, sc0,
      pObj, pCls, pLoc, accI, 80, 80, N0, C0, 0);
  stage1_kernel<<<NBATCH * C1, TPB, 0, stream>>>(p1, a1, bx, lb, sc1,
      pObj + pB1, pCls + pB1, pLoc + pB1, accI, 40, 40, N1, C1, 64);
  stage1_kernel<<<NBATCH * C2, TPB, 0, stream>>>(p2, a2, bx, lb, sc2,
      pObj + pB2, pCls + pB2, pLoc + pB2, accI, 20, 20, N2, C2, 128);

  stage2_kernel<<<NBATCH, TPB, 0, stream>>>(sc0, pObj, pCls, pLoc,
      accI, accF, topk, N0, C0, 0);
  stage2_kernel<<<NBATCH, TPB, 0, stream>>>(sc1, pObj + pB1, pCls + pB1, pLoc + pB1,
      accI, accF, topk, N1, C1, 64);
  stage2_kernel<<<NBATCH, TPB, 0, stream>>>(sc2, pObj + pB2, pCls + pB2, pLoc + pB2,
      accI, accF, topk, N2, C2, 128);

  finalize_kernel<<<1, 32, 0, stream>>>(accF, accI, topk, out);
}